// ConvBlock_64785286693635
// MI455X (gfx1250) — compile-verified
//
#include <hip/hip_runtime.h>
#include <hip/hip_bf16.h>
#include <stdint.h>

// CDNA5 / gfx1250: wave32, WMMA f32 16x16x4 (full f32 precision, matches reference dtype)
typedef __attribute__((ext_vector_type(2))) float v2f;
typedef __attribute__((ext_vector_type(4))) float v4f;
typedef __attribute__((ext_vector_type(8))) float v8f;

#define CIN 32
#define COUT 32
#define KCH 9   // weight taps staged in LDS per chunk: 9*16*32 float2 = 36,864 B

// ---------------------------------------------------------------------------
// Phase 1: gather-GEMM via V_WMMA_F32_16X16X4_F32.
// One wave owns one 16-row output tile (16 x 32).
//
// cin -> K-slot permutation (contraction is order-independent, A and B agree):
//   lanes 0-15  (hi=0): block b consumes cins {2b,   2b+1 }
//   lanes 16-31 (hi=1): block b consumes cins {16+2b,16+2b+1}
// => each lane's 16 gathered floats are CONTIGUOUS: 4 x global_load_b128.
// B-fragment cin-pair index = b + 8*hi; LDS staging layout is unchanged
// (consecutive-cin float2 pairs).
// ---------------------------------------------------------------------------
__global__ __launch_bounds__(256) void conv_wmma_kernel(
    const float* __restrict__ feat,          // (N, 32)
    const float* __restrict__ wgt,           // (K, 32, 32)
    const int*   __restrict__ nidx,          // (K, N)
    const unsigned char* __restrict__ nmask, // (K, N) bool
    float* __restrict__ out,                 // (N, 32) pre-BN
    int N, int K)
{
    // sW2[(kk*16 + cinPair)*32 + col] = { w[k][2*cinPair][col], w[k][2*cinPair+1][col] }
    __shared__ v2f sW2[KCH * 16 * 32];

    const int lane = threadIdx.x & 31;
    const int wave = threadIdx.x >> 5;
    const int colL = lane & 15;   // output column within 16-wide tile / A-row id
    const int hi   = lane >> 4;   // half-wave: cin group 0..15 vs 16..31

    const int tile = blockIdx.x * 8 + wave;
    const int r0   = tile * 16;                 // first row of this wave's tile
    const int rowA = r0 + colL;                 // row this lane gathers for A
    const bool rowOk = (rowA < N);
    const int rowAc  = rowOk ? rowA : 0;

    v8f acc_lo = {}; // cols 0..15
    v8f acc_hi = {}; // cols 16..31

    for (int k0 = 0; k0 < K; k0 += KCH) {
        const int cnt = (K - k0 < KCH) ? (K - k0) : KCH;

        __syncthreads(); // protect sW2 from previous chunk's readers
        // Cooperative stage: pair-transpose weights into LDS (float2 per 2 cin)
        for (int fi = threadIdx.x; fi < cnt * 512; fi += 256) {
            const int kk  = fi >> 9;        // tap within chunk
            const int rem = fi & 511;
            const int cp  = rem >> 5;       // cin pair 0..15
            const int col = rem & 31;       // cout 0..31
            const float* wk = wgt + (size_t)(k0 + kk) * (CIN * COUT);
            v2f w;
            w.x = wk[(2 * cp)     * COUT + col];
            w.y = wk[(2 * cp + 1) * COUT + col];
            sW2[fi] = w;
        }
        __syncthreads();

        if (r0 < N) { // wave-uniform branch: EXEC stays all-1s inside (WMMA requirement)
            for (int kk = 0; kk < cnt; ++kk) {
                const size_t kb = (size_t)(k0 + kk) * (size_t)N + (size_t)rowAc;
                const int   idx = nidx[kb];
                const float mf  = (rowOk && nmask[kb]) ? 1.0f : 0.0f;

                // 16 contiguous floats per lane: 4 x b128 gathers (64B-aligned)
                const v4f* fr4 = (const v4f*)(feat + (size_t)idx * CIN + 16 * hi);
                v4f f0 = fr4[0] * mf;
                v4f f1 = fr4[1] * mf;
                v4f f2 = fr4[2] * mf;
                v4f f3 = fr4[3] * mf;

                v2f a[8];
                a[0] = (v2f){f0.x, f0.y}; a[1] = (v2f){f0.z, f0.w};
                a[2] = (v2f){f1.x, f1.y}; a[3] = (v2f){f1.z, f1.w};
                a[4] = (v2f){f2.x, f2.y}; a[5] = (v2f){f2.z, f2.w};
                a[6] = (v2f){f3.x, f3.y}; a[7] = (v2f){f3.z, f3.w};

                // B frags: cin pair = b + 8*hi (matches permuted A layout)
                const v2f* bw = &sW2[(kk * 16 + 8 * hi) * 32];
                #pragma unroll
                for (int b = 0; b < 8; ++b) {
                    v2f b0 = bw[b * 32 + colL];        // cols 0..15
                    v2f b1 = bw[b * 32 + 16 + colL];   // cols 16..31
                    acc_lo = __builtin_amdgcn_wmma_f32_16x16x4_f32(
                        false, a[b], false, b0, (short)0, acc_lo, false, false);
                    acc_hi = __builtin_amdgcn_wmma_f32_16x16x4_f32(
                        false, a[b], false, b1, (short)0, acc_hi, false, false);
                }
            }
        }
    }

    if (r0 < N) {
        // C/D layout: VGPR r -> M = r + 8*hi; col = lane&15
        #pragma unroll
        for (int r = 0; r < 8; ++r) {
            const int orow = r0 + r + 8 * hi;
            if (orow < N) {
                out[(size_t)orow * COUT + colL]      = acc_lo[r];
                out[(size_t)orow * COUT + 16 + colL] = acc_hi[r];
            }
        }
    }
}

// ---------------------------------------------------------------------------
// Phase 2a: deterministic per-channel partial sums / sums-of-squares.
// Block b covers a contiguous row slab; fixed-order accumulation.
// ---------------------------------------------------------------------------
__global__ __launch_bounds__(256) void stats_partial_kernel(
    const float* __restrict__ out, float* __restrict__ part, int N, int nBlocks)
{
    __shared__ float sS[8][32];
    __shared__ float sQ[8][32];
    const int t = threadIdx.x;
    const int c = t & 31;
    const int g = t >> 5;

    const int rowsPer = (N + nBlocks - 1) / nBlocks;
    const int rStart  = blockIdx.x * rowsPer;
    const int rEnd    = (rStart + rowsPer < N) ? (rStart + rowsPer) : N;

    float s = 0.0f, q = 0.0f;
    for (int r = rStart + g; r < rEnd; r += 8) {
        const float v = out[(size_t)r * COUT + c];
        s += v;
        q += v * v;
    }
    sS[g][c] = s;
    sQ[g][c] = q;
    __syncthreads();

    if (t < 32) {
        float a = 0.0f;
        #pragma unroll
        for (int gg = 0; gg < 8; ++gg) a += sS[gg][t];
        part[(size_t)blockIdx.x * 64 + t] = a;
    } else if (t < 64) {
        const int cc = t - 32;
        float a = 0.0f;
        #pragma unroll
        for (int gg = 0; gg < 8; ++gg) a += sQ[gg][cc];
        part[(size_t)blockIdx.x * 64 + 32 + cc] = a;
    }
}

// Phase 2b: final deterministic reduction of block partials -> stats[64]
__global__ void stats_final_kernel(const float* __restrict__ part,
                                   float* __restrict__ stats, int nBlocks)
{
    const int t = threadIdx.x;
    if (t >= 64) return;
    float a = 0.0f;
    for (int b = 0; b < nBlocks; ++b) a += part[(size_t)b * 64 + t];
    stats[t] = a;
}

// ---------------------------------------------------------------------------
// Phase 3: in-place batchnorm (population var) + ReLU, float4 vectorized.
// ---------------------------------------------------------------------------
__global__ __launch_bounds__(256) void bn_relu_kernel(
    float* __restrict__ out, const float* __restrict__ stats,
    const float* __restrict__ gamma, const float* __restrict__ beta, int N)
{
    const size_t i = ((size_t)blockIdx.x * blockDim.x + threadIdx.x) * 4;
    const size_t total = (size_t)N * COUT;
    if (i >= total) return;

    float4 v = *reinterpret_cast<const float4*>(out + i);
    float r[4] = {v.x, v.y, v.z, v.w};
    const float invN = 1.0f / (float)N;
    const int c0 = (int)(i & 31);
    #pragma unroll
    for (int j = 0; j < 4; ++j) {
        const int c = c0 + j;
        const float mean = stats[c] * invN;
        const float var  = stats[32 + c] * invN - mean * mean;
        const float y = gamma[c] * (r[j] - mean) * rsqrtf(var + 1e-5f) + beta[c];
        r[j] = fmaxf(y, 0.0f);
    }
    float4 o = {r[0], r[1], r[2], r[3]};
    *reinterpret_cast<float4*>(out + i) = o;
}

// ---------------------------------------------------------------------------
extern "C" void kernel_launch(void* const* d_in, const int* in_sizes, int n_in,
                              void* d_out, int out_size, void* d_ws, size_t ws_size,
                              hipStream_t stream)
{
    const float* feat  = (const float*)d_in[0];
    const float* wgt   = (const float*)d_in[1];
    const float* gamma = (const float*)d_in[2];
    const float* beta  = (const float*)d_in[3];
    const int*   nidx  = (const int*)d_in[4];
    const unsigned char* nmask = (const unsigned char*)d_in[5];

    const int N = in_sizes[0] / CIN;            // 400,000
    const int K = in_sizes[1] / (CIN * COUT);   // 27

    float* out   = (float*)d_out;
    float* stats = (float*)d_ws;       // [64]: sum[32], sumsq[32]
    float* part  = stats + 64;         // [SB*64] partials

    const int tiles  = (N + 15) / 16;
    const int blk1   = (tiles + 7) / 8;         // 8 waves/block, 1 tile/wave
    conv_wmma_kernel<<<blk1, 256, 0, stream>>>(feat, wgt, nidx, nmask, out, N, K);

    const int SB = 256;
    stats_partial_kernel<<<SB, 256, 0, stream>>>(out, part, N, SB);
    stats_final_kernel<<<1, 64, 0, stream>>>(part, stats, SB);

    const size_t total4 = ((size_t)N * COUT) / 4;
    const int blk3 = (int)((total4 + 255) / 256);
    bn_relu_kernel<<<blk3, 256, 0, stream>>>(out, stats, gamma, beta, N);
}